// GATAgent_2353642078262
// MI455X (gfx1250) — compile-verified
//
#include <hip/hip_runtime.h>

typedef float v2f __attribute__((ext_vector_type(2)));
typedef float v4f __attribute__((ext_vector_type(4)));
typedef float v8f __attribute__((ext_vector_type(8)));

#define NODE_IN 64
#define EDGE_IN 32
#define HIDDEN  128
#define NB      32
#define NN      256

// -------- Kernel 1: h[m][n] = sum_k node[m][k] * fcw[n][k]   (M=8192,K=64,N=128)
// grid 512 blocks (16 M-rows each), 256 threads = 8 waves (one 16-col N-tile per wave)
__global__ __launch_bounds__(256) void gat_h_kernel(const float* __restrict__ node,
                                                    const float* __restrict__ fcw,
                                                    float* __restrict__ h)
{
    __shared__ float nodeT[16][66];   // padded: stride 66 avoids bank conflicts
    const int tid = threadIdx.x;
    const int m0  = blockIdx.x * 16;

    for (int idx = tid; idx < 16 * NODE_IN; idx += 256)
        nodeT[idx >> 6][idx & 63] = node[(size_t)m0 * NODE_IN + idx];
    __syncthreads();

    const int wave = tid >> 5, lane = tid & 31;
    const int half = lane >> 4, lm = lane & 15;
    const int nbase = wave * 16;

    v8f acc = {};
    for (int k0 = 0; k0 < NODE_IN; k0 += 4) {
        const int ka = k0 + 2 * half;
        v2f a, b;
        a.x = nodeT[lm][ka];
        a.y = nodeT[lm][ka + 1];
        // B[k][n] = fcw[n][k]   (fcw is [128][64], tiny & cache-resident)
        b.x = fcw[(nbase + lm) * NODE_IN + ka];
        b.y = fcw[(nbase + lm) * NODE_IN + ka + 1];
        acc = __builtin_amdgcn_wmma_f32_16x16x4_f32(false, a, false, b,
                                                    (short)0, acc, false, false);
    }
    #pragma unroll
    for (int v = 0; v < 8; ++v) {
        const int m = m0 + v + 8 * half;          // D layout: M = v + 8*(lane/16)
        h[(size_t)m * HIDDEN + nbase + lm] = acc[v];
    }
}

// -------- Kernel 2: per (b, 16-row i-tile): s_e stream + softmax + alpha@h
// grid 512 blocks (b*16 + itile), 256 threads = 8 waves
__global__ __launch_bounds__(256) void gat_attn_kernel(const float* __restrict__ edge,
                                                       const float* __restrict__ attn,
                                                       const float* __restrict__ h,
                                                       float* __restrict__ out)
{
    __shared__ float hS[NN][HIDDEN + 4];      // stride 132: kills the stride-128 bank conflict
    __shared__ float eS[16][NN + 1];          // e rows -> alpha rows (stride 257)
    __shared__ float sJ[NN];
    __shared__ float sI[16];
    __shared__ float aS[2 * HIDDEN + EDGE_IN];
    __shared__ float red[16][16];
    __shared__ float rowM[16];
    __shared__ float rowInv[16];

    const int tid   = threadIdx.x;
    const int b     = blockIdx.x >> 4;
    const int itile = blockIdx.x & 15;
    const int i0    = itile * 16;

    // ---- load h_b (256x128 f32 = 128 KB) into LDS, plus attention vector
    const float* hb = h + (size_t)b * NN * HIDDEN;
    for (int idx = tid * 4; idx < NN * HIDDEN; idx += 256 * 4) {
        v4f v = *(const v4f*)(hb + idx);
        const int r = idx >> 7, c = idx & 127;
        hS[r][c] = v.x; hS[r][c + 1] = v.y; hS[r][c + 2] = v.z; hS[r][c + 3] = v.w;
    }
    if (tid < 2 * HIDDEN + EDGE_IN) aS[tid] = attn[tid];
    __syncthreads();

    // ---- s_j for all 256 j (one per thread); s_i for the 16 local rows
    {
        float s = 0.f;
        for (int n = 0; n < HIDDEN; ++n) s += hS[tid][n] * aS[HIDDEN + n];
        sJ[tid] = s;
        if (tid < 16) {
            float si = 0.f;
            for (int n = 0; n < HIDDEN; ++n) si += hS[i0 + tid][n] * aS[n];
            sI[tid] = si;
        }
    }
    __syncthreads();

    // ---- stream edge_feats (the 268 MB of the problem): 8 lanes per 32-float dot,
    // fully coalesced 128B groups, non-temporal so L2 keeps h instead.
    {
        const float* eb = edge + (size_t)b * NN * NN * EDGE_IN;
        const int sub = tid & 7;
        const int c   = sub * 4;
        const float ae0 = aS[2 * HIDDEN + c + 0], ae1 = aS[2 * HIDDEN + c + 1];
        const float ae2 = aS[2 * HIDDEN + c + 2], ae3 = aS[2 * HIDDEN + c + 3];
        for (int pbase = 0; pbase < 16 * NN; pbase += 32) {
            const int p = pbase + (tid >> 3);
            const int i = p >> 8, j = p & 255;
            const v4f* gp = (const v4f*)(eb + ((size_t)(i0 + i) * NN + j) * EDGE_IN + c);
            v4f ev = __builtin_nontemporal_load(gp);
            float partial = ev.x * ae0 + ev.y * ae1 + ev.z * ae2 + ev.w * ae3;
            partial += __shfl_xor(partial, 4, 32);
            partial += __shfl_xor(partial, 2, 32);
            partial += __shfl_xor(partial, 1, 32);
            if (sub == 0) eS[i][j] = partial + sI[i] + sJ[j];
        }
    }
    __syncthreads();

    // ---- softmax over j (256) per row: 16 threads/row, strided-16 j mapping
    {
        const int r = tid >> 4, s = tid & 15;
        float m = -__builtin_inff();
        for (int jj = 0; jj < 16; ++jj) m = fmaxf(m, eS[r][s + jj * 16]);
        red[r][s] = m;
        __syncthreads();
        if (s == 0) {
            float mm = red[r][0];
            for (int q = 1; q < 16; ++q) mm = fmaxf(mm, red[r][q]);
            rowM[r] = mm;
        }
        __syncthreads();
        const float rm = rowM[r];
        float sum = 0.f;
        for (int jj = 0; jj < 16; ++jj) {
            const int j = s + jj * 16;
            const float ex = __expf(eS[r][j] - rm);
            eS[r][j] = ex;
            sum += ex;
        }
        red[r][s] = sum;
        __syncthreads();
        if (s == 0) {
            float ss = 0.f;
            for (int q = 0; q < 16; ++q) ss += red[r][q];
            rowInv[r] = 1.0f / ss;
        }
        __syncthreads();
        const float inv = rowInv[r];
        for (int jj = 0; jj < 16; ++jj) eS[r][s + jj * 16] *= inv;
    }
    __syncthreads();

    // ---- h' tile: alpha(16x256) @ h_b(256x128) via f32 WMMA, 8 waves = 8 N-tiles
    {
        const int wave = tid >> 5, lane = tid & 31;
        const int half = lane >> 4, lm = lane & 15;
        const int nbase = wave * 16;
        v8f acc = {};
        for (int k0 = 0; k0 < NN; k0 += 4) {
            const int ka = k0 + 2 * half;
            v2f a, bb;
            a.x  = eS[lm][ka];                  // A: M=lane%16, K=2*(lane/16)+v
            a.y  = eS[lm][ka + 1];
            bb.x = hS[ka][nbase + lm];          // B: K rows of h, N=lane%16
            bb.y = hS[ka + 1][nbase + lm];
            acc = __builtin_amdgcn_wmma_f32_16x16x4_f32(false, a, false, bb,
                                                        (short)0, acc, false, false);
        }
        float* ob = out + ((size_t)b * NN + i0) * HIDDEN;
        #pragma unroll
        for (int v = 0; v < 8; ++v)
            ob[(size_t)(v + 8 * half) * HIDDEN + nbase + lm] = acc[v];
    }
}

extern "C" void kernel_launch(void* const* d_in, const int* in_sizes, int n_in,
                              void* d_out, int out_size, void* d_ws, size_t ws_size,
                              hipStream_t stream) {
    const float* node = (const float*)d_in[0];   // (32,256,64)
    const float* edge = (const float*)d_in[1];   // (32,256,256,32)
    const float* fcw  = (const float*)d_in[2];   // (128,64)
    const float* attn = (const float*)d_in[3];   // (1,288)
    float* h = (float*)d_ws;                     // 32*256*128 f32 = 4 MB scratch

    gat_h_kernel<<<512, 256, 0, stream>>>(node, fcw, h);
    gat_attn_kernel<<<512, 256, 0, stream>>>(edge, attn, h, (float*)d_out);
}